// GCN_80255758893370
// MI455X (gfx1250) — compile-verified
//
#include <hip/hip_runtime.h>
#include <hip/hip_bf16.h>

typedef __attribute__((ext_vector_type(2))) float v2f;
typedef __attribute__((ext_vector_type(8))) float v8f;

// ---------------------------------------------------------------------------
// Utility fill
// ---------------------------------------------------------------------------
__global__ void fill_kernel(float* __restrict__ p, float v, size_t n) {
    size_t i = (size_t)blockIdx.x * blockDim.x + threadIdx.x;
    if (i < n) p[i] = v;
}

// ---------------------------------------------------------------------------
// Pack W[K,N] -> P with consecutive row pairs interleaved by column, padded
// to NPAD columns: P[(k>>1)*2*NPAD + 2*c + (k&1)] = W[k][c]  (0 for c >= N).
// Lets the GEMM load each B fragment (two K-adjacent rows, same col) as b64.
// ---------------------------------------------------------------------------
__global__ void pack_w_kernel(const float* __restrict__ W, float* __restrict__ P,
                              int K, int N, int NPAD) {
    int idx = blockIdx.x * blockDim.x + threadIdx.x;
    if (idx >= K * NPAD) return;
    int k = idx / NPAD;
    int c = idx - k * NPAD;
    float v = (c < N) ? W[(size_t)k * N + c] : 0.0f;
    P[(size_t)(k >> 1) * 2 * NPAD + 2 * c + (k & 1)] = v;
}

// ---------------------------------------------------------------------------
// Degree accumulation (deg buffer pre-filled with 1.0 for the self loop)
// ---------------------------------------------------------------------------
__global__ void degree_kernel(const int* __restrict__ dst, float* __restrict__ deg, int E) {
    int e = blockIdx.x * blockDim.x + threadIdx.x;
    if (e < E) atomicAdd(&deg[dst[e]], 1.0f);
}

__global__ void rsqrt_kernel(float* __restrict__ deg, int N) {
    int i = blockIdx.x * blockDim.x + threadIdx.x;
    if (i < N) {
        float d = deg[i];
        deg[i] = (d > 0.0f) ? rsqrtf(d) : 0.0f;
    }
}

// ---------------------------------------------------------------------------
// fp32 WMMA GEMM: Out[M,Nout] = A[M,K] @ W[K,Nout], W pre-packed to NPAD cols.
// One wave computes TWO 16-row tiles x NT*16 columns (V_WMMA_F32_16X16X4_F32).
// __launch_bounds__(128, 1): allow full VGPR budget so the 2*NT v8f
// accumulators (128 VGPRs at NT=8) never spill to scratch.
// ---------------------------------------------------------------------------
template <int NT, int NPAD>
__global__ __launch_bounds__(128, 1) void wmma_gemm_f32(
        const float* __restrict__ A, const float* __restrict__ Wp,
        float* __restrict__ Out, int M, int K, int Nout) {
    const int lane = threadIdx.x & 31;
    const int wave = threadIdx.x >> 5;
    const int nPairTiles = (M + 31) >> 5;
    const int pairTile = blockIdx.x * (blockDim.x >> 5) + wave;
    if (pairTile >= nPairTiles) return;          // wave-uniform exit

    const int m0    = pairTile * 32 + (lane & 15);
    const int koff  = (lane >> 4) << 1;          // 0 or 2 (A k offset)
    const int koff2 = (lane >> 4);               // 0 or 1 (packed row-pair)
    const int n     = lane & 15;

    // second tile valid? (M is a multiple of 16, so this is wave-uniform)
    const bool hasSecond = (pairTile * 32 + 16) < M;

    v8f acc0[NT], acc1[NT];
#pragma unroll
    for (int t = 0; t < NT; ++t) {
        v8f z = {0.f, 0.f, 0.f, 0.f, 0.f, 0.f, 0.f, 0.f};
        acc0[t] = z;
        acc1[t] = z;
    }

    const float* A0 = A + (size_t)m0 * K;
    const float* A1 = hasSecond ? (A0 + (size_t)16 * K) : A0;  // alias when absent

    for (int k = 0; k < K; k += 4) {
        v2f a0 = *(const v2f*)(A0 + k + koff);
        v2f a1 = *(const v2f*)(A1 + k + koff);
        const float* Wrow = Wp + (size_t)((k >> 1) + koff2) * (2 * NPAD);
#pragma unroll
        for (int t = 0; t < NT; ++t) {
            v2f b = *(const v2f*)(Wrow + 2 * (t * 16 + n));
            acc0[t] = __builtin_amdgcn_wmma_f32_16x16x4_f32(
                false, a0, false, b, (short)0, acc0[t], false, false);
            acc1[t] = __builtin_amdgcn_wmma_f32_16x16x4_f32(
                false, a1, false, b, (short)0, acc1[t], false, false);
        }
    }

    const int rbase0 = pairTile * 32 + ((lane >> 4) << 3);
    const int rbase1 = rbase0 + 16;
#pragma unroll
    for (int t = 0; t < NT; ++t) {
        int col = t * 16 + n;
        if (col < Nout) {
#pragma unroll
            for (int i = 0; i < 8; ++i)
                Out[(size_t)(rbase0 + i) * Nout + col] = acc0[t][i];
            if (hasSecond) {
#pragma unroll
                for (int i = 0; i < 8; ++i)
                    Out[(size_t)(rbase1 + i) * Nout + col] = acc1[t][i];
            }
        }
    }
}

// ---------------------------------------------------------------------------
// Edge scatter: acc[dst] += H[src] * dinv[src]*dinv[dst]
// One wave per edge; each lane handles a float4 feature chunk.
// ---------------------------------------------------------------------------
__global__ void scatter_kernel(const float* __restrict__ H, const int* __restrict__ src,
                               const int* __restrict__ dst, const float* __restrict__ dinv,
                               float* __restrict__ acc, int E, int F) {
    const int lane = threadIdx.x & 31;
    const int wave = threadIdx.x >> 5;
    const int e = blockIdx.x * (blockDim.x >> 5) + wave;
    if (e >= E) return;
    const int f0 = lane << 2;
    if (f0 >= F) return;
    const int s = src[e];
    const int d = dst[e];
    const float norm = dinv[s] * dinv[d];
    const float4 h = *(const float4*)(H + (size_t)s * F + f0);
    float* out = acc + (size_t)d * F + f0;
    atomicAdd(out + 0, h.x * norm);
    atomicAdd(out + 1, h.y * norm);
    atomicAdd(out + 2, h.z * norm);
    atomicAdd(out + 3, h.w * norm);
}

// ---------------------------------------------------------------------------
// acc = relu(acc + h * dinv[n]^2 + b[f])   (self-loop contribution fused)
// F = 1 << logF
// ---------------------------------------------------------------------------
__global__ void bias_relu_self_kernel(float* __restrict__ acc, const float* __restrict__ h,
                                      const float* __restrict__ dinv, const float* __restrict__ b,
                                      int N, int logF) {
    size_t i = (size_t)blockIdx.x * blockDim.x + threadIdx.x;
    if (i >= ((size_t)N << logF)) return;
    int node = (int)(i >> logF);
    int f = (int)(i & ((1u << logF) - 1));
    float dn = dinv[node];
    float v = acc[i] + h[i] * dn * dn + b[f];
    acc[i] = v > 0.0f ? v : 0.0f;
}

// ---------------------------------------------------------------------------
// Final: logits = acc2 + h2*dinv^2 + b2 ; out = log_softmax(logits) per node.
// One wave per node; C <= 64 classes (lane covers f and f+32).
// ---------------------------------------------------------------------------
__global__ void logsoftmax_kernel(const float* __restrict__ acc2, const float* __restrict__ h2,
                                  const float* __restrict__ dinv, const float* __restrict__ b2,
                                  float* __restrict__ out, int N, int C) {
    const int lane = threadIdx.x & 31;
    const int node = (blockIdx.x * blockDim.x + threadIdx.x) >> 5;
    if (node >= N) return;
    const float dn = dinv[node];
    const float sn = dn * dn;
    const int f0 = lane, f1 = lane + 32;
    const size_t base = (size_t)node * C;
    float v0 = -3.4e38f, v1 = -3.4e38f;
    if (f0 < C) v0 = acc2[base + f0] + h2[base + f0] * sn + b2[f0];
    if (f1 < C) v1 = acc2[base + f1] + h2[base + f1] * sn + b2[f1];
    float mx = fmaxf(v0, v1);
#pragma unroll
    for (int off = 16; off > 0; off >>= 1) mx = fmaxf(mx, __shfl_xor(mx, off, 32));
    float s = 0.0f;
    if (f0 < C) s += __expf(v0 - mx);
    if (f1 < C) s += __expf(v1 - mx);
#pragma unroll
    for (int off = 16; off > 0; off >>= 1) s += __shfl_xor(s, off, 32);
    const float lse = __logf(s) + mx;
    if (f0 < C) out[base + f0] = v0 - lse;
    if (f1 < C) out[base + f1] = v1 - lse;
}

// ---------------------------------------------------------------------------
// Host-side launch
// ---------------------------------------------------------------------------
extern "C" void kernel_launch(void* const* d_in, const int* in_sizes, int n_in,
                              void* d_out, int out_size, void* d_ws, size_t ws_size,
                              hipStream_t stream) {
    const float* x  = (const float*)d_in[0];   // [N, 256]
    const int*   ei = (const int*)d_in[1];     // [2, E]
    const float* W1 = (const float*)d_in[2];   // [256, 128]
    const float* b1 = (const float*)d_in[3];   // [128]
    const float* W2 = (const float*)d_in[4];   // [128, 40]
    const float* b2 = (const float*)d_in[5];   // [40]
    float* out = (float*)d_out;

    const int F_IN  = 256;
    const int HID   = 128;    // = 1 << 7
    const int NCLS  = 40;
    const int NPAD2 = 48;
    const int N = in_sizes[0] / F_IN;          // 50000
    const int E = in_sizes[1] / 2;             // 800000
    const int* src = ei;
    const int* dst = ei + E;

    // Workspace carve-up (all fp32), ~67.7 MB total
    float* dinv = (float*)d_ws;                       // N
    float* h1   = dinv + ((N + 63) & ~63);            // N*HID
    float* acc1 = h1 + (size_t)N * HID;               // N*HID
    float* h2   = acc1 + (size_t)N * HID;             // N*NCLS
    float* acc2 = h2 + (size_t)N * NCLS;              // N*NCLS
    float* W1p  = acc2 + (size_t)N * NCLS;            // F_IN*HID  packed
    float* W2p  = W1p + (size_t)F_IN * HID;           // HID*NPAD2 packed

    // --- init: deg=1 (self loop), acc1=0, acc2=0; pack weights ---
    {
        size_t n1 = N;
        fill_kernel<<<(unsigned)((n1 + 255) / 256), 256, 0, stream>>>(dinv, 1.0f, n1);
        size_t n2 = (size_t)N * HID;
        fill_kernel<<<(unsigned)((n2 + 255) / 256), 256, 0, stream>>>(acc1, 0.0f, n2);
        size_t n3 = (size_t)N * NCLS;
        fill_kernel<<<(unsigned)((n3 + 255) / 256), 256, 0, stream>>>(acc2, 0.0f, n3);
        pack_w_kernel<<<(F_IN * HID + 255) / 256, 256, 0, stream>>>(W1, W1p, F_IN, HID, HID);
        pack_w_kernel<<<(HID * NPAD2 + 255) / 256, 256, 0, stream>>>(W2, W2p, HID, NCLS, NPAD2);
    }

    // --- degree + rsqrt ---
    degree_kernel<<<(E + 255) / 256, 256, 0, stream>>>(dst, dinv, E);
    rsqrt_kernel<<<(N + 255) / 256, 256, 0, stream>>>(dinv, N);

    // --- layer 1 GEMM: h1 = x @ W1 (fp32 WMMA, 2 row tiles/wave) ---
    {
        const int pairTiles = (N + 31) / 32;
        const int wavesPerBlock = 4;  // 128 threads
        const int blocks = (pairTiles + wavesPerBlock - 1) / wavesPerBlock;
        wmma_gemm_f32<8, 128><<<blocks, wavesPerBlock * 32, 0, stream>>>(x, W1p, h1, N, F_IN, HID);
    }

    // --- layer 1 scatter over edges ---
    {
        const int edgesPerBlock = 4;  // 4 waves of 32
        const int blocks = (E + edgesPerBlock - 1) / edgesPerBlock;
        scatter_kernel<<<blocks, edgesPerBlock * 32, 0, stream>>>(h1, src, dst, dinv, acc1, E, HID);
    }

    // --- fused self-loop + bias + relu ---
    {
        size_t total = (size_t)N * HID;
        bias_relu_self_kernel<<<(unsigned)((total + 255) / 256), 256, 0, stream>>>(
            acc1, h1, dinv, b1, N, 7);
    }

    // --- layer 2 GEMM: h2 = relu_h @ W2 (fp32 WMMA, padded to 48 cols) ---
    {
        const int pairTiles = (N + 31) / 32;
        const int wavesPerBlock = 4;
        const int blocks = (pairTiles + wavesPerBlock - 1) / wavesPerBlock;
        wmma_gemm_f32<3, 48><<<blocks, wavesPerBlock * 32, 0, stream>>>(acc1, W2p, h2, N, HID, NCLS);
    }

    // --- layer 2 scatter ---
    {
        const int edgesPerBlock = 4;
        const int blocks = (E + edgesPerBlock - 1) / edgesPerBlock;
        scatter_kernel<<<blocks, edgesPerBlock * 32, 0, stream>>>(h2, src, dst, dinv, acc2, E, NCLS);
    }

    // --- final self-loop + bias + log_softmax ---
    {
        const int wavesPerBlock = 8;  // 256 threads
        const int blocks = (N + wavesPerBlock - 1) / wavesPerBlock;
        logsoftmax_kernel<<<blocks, wavesPerBlock * 32, 0, stream>>>(
            acc2, h2, dinv, b2, out, N, NCLS);
    }
}